// MIEstimator_20289425506547
// MI455X (gfx1250) — compile-verified
//
#include <hip/hip_runtime.h>
#include <math.h>

typedef float v2f __attribute__((ext_vector_type(2)));
typedef float v8f __attribute__((ext_vector_type(8)));

#define NB   16
#define LL   128
#define DD   400          // 3*L + nb
#define HID  1024
#define TT   9
#define TM1  8
#define BBZ  1024
#define NN   (BBZ*TM1)    // 8192 rows
#define MM   (2*NN)       // 16384 joint rows

// ---------------- preprocessing ----------------

__global__ __launch_bounds__(256) void mie_intv(const float* __restrict__ tgt,
                                                int* __restrict__ intv) {
  int n = blockIdx.x * 256 + threadIdx.x;
  if (n >= NN) return;
  const float* t = tgt + (size_t)n * NB;
  float best = t[0]; int a = 0;
#pragma unroll
  for (int b = 1; b < NB; ++b) { float v = t[b]; if (v > best) { best = v; a = b; } }
  intv[n] = a;
}

// stable counting sort by intervention target (16 keys)
__global__ __launch_bounds__(32) void mie_sort(const int* __restrict__ intv,
                                               int* __restrict__ order,
                                               int* __restrict__ base) {
  __shared__ int cnt[NB];
  __shared__ int sbase[NB + 1];
  int b = threadIdx.x;
  if (b < NB) {
    int c = 0;
    for (int n = 0; n < NN; ++n) c += (intv[n] == b);
    cnt[b] = c;
  }
  __syncthreads();
  if (b == 0) {
    int s = 0;
    for (int i = 0; i < NB; ++i) { sbase[i] = s; s += cnt[i]; }
    sbase[NB] = s;
    for (int i = 0; i <= NB; ++i) base[i] = sbase[i];
  }
  __syncthreads();
  if (b < NB) {
    int p = sbase[b];
    for (int n = 0; n < NN; ++n) if (intv[n] == b) order[p++] = n;
  }
}

// build inp_sel row (2n) and the shared parts of inp_alt row (2n+1)
__global__ __launch_bounds__(128) void mie_build(const float* __restrict__ z,
      const float* __restrict__ tp, const float* __restrict__ gn,
      const float* __restrict__ gp, const float* __restrict__ bu,
      const int* __restrict__ order, const int* __restrict__ intv,
      float* __restrict__ X, float* __restrict__ lm_s, float* __restrict__ z1_s) {
  const int n = blockIdx.x;    // sorted row
  const int l = threadIdx.x;   // latent 0..127
  const int orig = order[n];
  const int c = intv[orig];
  const float* xg = gn + ((size_t)orig * LL + l) * NB;
  const float* xt = tp + l * NB;
  float xv[NB];
  float mx = -3.402823466e38f; int am = 0;
#pragma unroll
  for (int b = 0; b < NB; ++b) {
    float t = xt[b] + xg[b];   // TAU = 1
    xv[b] = t;
    if (t > mx) { mx = t; am = b; }
  }
  float den = 0.f;
#pragma unroll
  for (int b = 0; b < NB; ++b) den += expf(xv[b] - mx);
  const float s    = 1.f / den;            // y_soft at argmax
  const float tval = (1.f - s) + s;        // straight-through forward value
  const float gv = ((bu[((size_t)orig * NB + am) * NB + c] < gp[am * NB + c]) ? 1.f : 0.f)
                 - ((am == c) ? 1.f : 0.f);
  const float lm  = tval * gv;
  const float lma = fabsf(lm);
  const int batch = orig / TM1, tt = orig % TM1;
  const float* zr = z + ((size_t)batch * TT + tt) * LL;
  const float z0v = zr[l];
  const float z1v = zr[LL + l];
  const size_t r0 = (size_t)(2 * n) * DD, r1 = r0 + DD;
  X[r0 + l] = z0v;               X[r1 + l] = z0v;
  X[r0 + LL + l] = z1v * lma;
  X[r0 + 2 * LL + l] = lm;       X[r1 + 2 * LL + l] = lm;
  lm_s[(size_t)n * LL + l] = lm;
  z1_s[(size_t)n * LL + l] = z1v;
  if (l < NB) {
    const float oh = (l == c) ? 1.f : 0.f;
    X[r0 + 3 * LL + l] = oh;
    X[r1 + 3 * LL + l] = oh;
  }
}

// stable argsort(intv_s + perm_noise): rank within contiguous block
__global__ __launch_bounds__(256) void mie_alt(const float* __restrict__ pn,
      const int* __restrict__ base, const int* __restrict__ order,
      const int* __restrict__ intv, int* __restrict__ alt) {
  const int i = blockIdx.x * 256 + threadIdx.x;
  if (i >= NN) return;
  const int b = intv[order[i]];
  const int lo = base[b], hi = base[b + 1];
  const float k = pn[i];
  int rank = 0;
  for (int j = lo; j < hi; ++j) {
    float kj = pn[j];
    rank += (kj < k) || ((kj == k) && (j < i));
  }
  alt[lo + rank] = i;
}

__global__ __launch_bounds__(128) void mie_fill_alt(const int* __restrict__ alt,
      const float* __restrict__ lm_s, const float* __restrict__ z1_s,
      float* __restrict__ X) {
  const int n = blockIdx.x;
  const int l = threadIdx.x;
  const int a = alt[n];
  const float lm  = lm_s[(size_t)n * LL + l];
  const float z1v = z1_s[(size_t)n * LL + l];
  const float z1a = z1_s[(size_t)a * LL + l];
  X[(size_t)(2 * n + 1) * DD + LL + l] = ((lm == -1.f) ? z1a : z1v) * fabsf(lm);
}

// ---------------- fp32 WMMA GEMM: C[M,N] = A[M,K] @ B[K,N] ----------------
// 64x128 block tile, 8 waves, 4 accumulator tiles/wave, register double-buffered.

#define BM 64
#define BN 128
#define BK 16
#define BNP (BN + 16)     // row stride mod 64 == 16 -> conflict-free B-frag reads

__global__ __launch_bounds__(256) void mie_gemm(const float* __restrict__ A,
                                                const float* __restrict__ B,
                                                float* __restrict__ C,
                                                int M, int N, int K) {
  __shared__ float As[BM][BK + 1];
  __shared__ float Bs[BK][BNP];
  const int tid  = threadIdx.x;
  const int lane = tid & 31;
  const int wave = tid >> 5;          // 8 waves
  const int wr   = wave & 3;          // row tile 0..3   (x16 -> 64 rows)
  const int wc   = (wave >> 2) << 2;  // col tile base 0 or 4 (x16 -> 128 cols)
  const int m16  = lane & 15;
  const int kh   = lane >> 4;         // K-pair select (ISA A/B frag layout)
  const int bm = blockIdx.y * BM;
  const int bn = blockIdx.x * BN;

  v8f acc[4];
#pragma unroll
  for (int t = 0; t < 4; ++t) acc[t] = (v8f){0.f,0.f,0.f,0.f,0.f,0.f,0.f,0.f};

  const int ar = tid >> 2, ac = (tid & 3) << 2;    // A tile 64x16: 1 float4/thread
  const int br = tid >> 5, bc = (tid & 31) << 2;   // B tile 16x128: rows br, br+8

  const float* Aptr  = A + (size_t)(bm + ar) * K + ac;
  const float* Bptr0 = B + (size_t)br * N + (bn + bc);
  const float* Bptr1 = B + (size_t)(br + 8) * N + (bn + bc);
  const size_t bstep = (size_t)BK * N;

  // prologue prefetch (tile 0)
  float4 a_reg  = *(const float4*)Aptr;
  float4 b0_reg = *(const float4*)Bptr0;
  float4 b1_reg = *(const float4*)Bptr1;

  for (int k0 = 0; k0 < K; k0 += BK) {
    As[ar][ac+0]=a_reg.x;  As[ar][ac+1]=a_reg.y;  As[ar][ac+2]=a_reg.z;  As[ar][ac+3]=a_reg.w;
    Bs[br][bc+0]=b0_reg.x; Bs[br][bc+1]=b0_reg.y; Bs[br][bc+2]=b0_reg.z; Bs[br][bc+3]=b0_reg.w;
    Bs[br+8][bc+0]=b1_reg.x; Bs[br+8][bc+1]=b1_reg.y; Bs[br+8][bc+2]=b1_reg.z; Bs[br+8][bc+3]=b1_reg.w;
    __syncthreads();
    if (k0 + BK < K) {                 // uniform branch; prefetch next tile
      Aptr += BK; Bptr0 += bstep; Bptr1 += bstep;
      a_reg  = *(const float4*)Aptr;
      b0_reg = *(const float4*)Bptr0;
      b1_reg = *(const float4*)Bptr1;
    }
#pragma unroll
    for (int kk = 0; kk < BK; kk += 4) {
      v2f a;
      a.x = As[wr*16 + m16][kk + 2*kh];
      a.y = As[wr*16 + m16][kk + 2*kh + 1];
#pragma unroll
      for (int t = 0; t < 4; ++t) {
        v2f b;
        b.x = Bs[kk + 2*kh][(wc + t)*16 + m16];
        b.y = Bs[kk + 2*kh + 1][(wc + t)*16 + m16];
        acc[t] = __builtin_amdgcn_wmma_f32_16x16x4_f32(false, a, false, b,
                                                       (short)0, acc[t], false, false);
      }
    }
    __syncthreads();
  }
  const int crow = bm + wr*16 + 8*kh;  // C/D layout: VGPR i -> row i + 8*(lane/16)
#pragma unroll
  for (int t = 0; t < 4; ++t) {
    const int cc = bn + (wc + t)*16 + m16;
#pragma unroll
    for (int i = 0; i < 8; ++i)
      C[(size_t)(crow + i) * N + cc] = acc[t][i];
  }
}

// ---------------- bias + LayerNorm + SiLU (row of 1024) ----------------

__global__ __launch_bounds__(256) void mie_ln_silu(float* __restrict__ Y,
      const float* __restrict__ bias, const float* __restrict__ g,
      const float* __restrict__ be) {
  __shared__ float red[256];
  const int tid = threadIdx.x;
  float* y = Y + (size_t)blockIdx.x * HID;
  float v[4]; float s = 0.f;
#pragma unroll
  for (int i = 0; i < 4; ++i) { int c = tid + i*256; v[i] = y[c] + bias[c]; s += v[i]; }
  red[tid] = s; __syncthreads();
  for (int o = 128; o > 0; o >>= 1) { if (tid < o) red[tid] += red[tid+o]; __syncthreads(); }
  const float mean = red[0] * (1.f / HID);
  __syncthreads();
  float sq = 0.f;
#pragma unroll
  for (int i = 0; i < 4; ++i) { float d = v[i] - mean; sq += d*d; }
  red[tid] = sq; __syncthreads();
  for (int o = 128; o > 0; o >>= 1) { if (tid < o) red[tid] += red[tid+o]; __syncthreads(); }
  const float inv = rsqrtf(red[0] * (1.f / HID) + 1e-5f);
#pragma unroll
  for (int i = 0; i < 4; ++i) {
    int c = tid + i*256;
    float x = (v[i] - mean) * inv * g[c] + be[c];
    y[c] = x / (1.f + expf(-x));       // SiLU
  }
}

// ---------------- final projection H->1 : one wave per row ----------------

__global__ __launch_bounds__(256) void mie_rowdot(const float* __restrict__ Hm,
      const float* __restrict__ w3, float* __restrict__ out) {
  const int row  = blockIdx.x * 8 + (threadIdx.x >> 5);
  const int lane = threadIdx.x & 31;
  const float* r = Hm + (size_t)row * HID;
  float s = 0.f;
  for (int c = lane; c < HID; c += 32) s += r[c] * w3[c];
#pragma unroll
  for (int o = 16; o > 0; o >>= 1) s += __shfl_xor(s, o, 32);
  if (lane == 0) out[row] = s;
}

// ---------------- losses ----------------

__global__ __launch_bounds__(256) void mie_loss(const float* __restrict__ mo,
      const float* __restrict__ zo, float* __restrict__ out) {
  __shared__ float r0[256], r1[256], r2[256];
  const int tid = threadIdx.x;
  float s_lm = 0.f, s_sq = 0.f, s_lz = 0.f;
  for (int n = tid; n < NN; n += 256) {
    float a = mo[2*n], b = mo[2*n+1];
    float mx = fmaxf(a, b);
    float lse = mx + logf(expf(a - mx) + expf(b - mx));
    s_lm += lse - a;
    s_sq += a*a + b*b;
    float za = zo[2*n], zb = zo[2*n+1];
    float zmx = fmaxf(za, zb);
    float zlse = zmx + logf(expf(za - zmx) + expf(zb - zmx));
    s_lz += zlse - zb;                 // -log_softmax[:, -1]
  }
  r0[tid]=s_lm; r1[tid]=s_sq; r2[tid]=s_lz; __syncthreads();
  for (int o = 128; o > 0; o >>= 1) {
    if (tid < o) { r0[tid]+=r0[tid+o]; r1[tid]+=r1[tid+o]; r2[tid]+=r2[tid+o]; }
    __syncthreads();
  }
  if (tid == 0) {
    out[0] = r0[0] / (float)NN + 1e-3f * (r1[0] / (float)(2 * NN));
    out[1] = r2[0] / (float)NN;
  }
}

// ---------------- driver ----------------

extern "C" void kernel_launch(void* const* d_in, const int* in_sizes, int n_in,
                              void* d_out, int out_size, void* d_ws, size_t ws_size,
                              hipStream_t stream) {
  const float* z_sample = (const float*)d_in[0];
  const float* target   = (const float*)d_in[1];
  const float* tparams  = (const float*)d_in[2];
  const float* gprobs   = (const float*)d_in[3];
  const float* gnoise   = (const float*)d_in[4];
  const float* bern     = (const float*)d_in[5];
  const float* pnoise   = (const float*)d_in[6];
  const float* miw[9]; const float* exw[9];
  for (int i = 0; i < 9; ++i) { miw[i] = (const float*)d_in[7 + i]; exw[i] = (const float*)d_in[16 + i]; }

  char* ws = (char*)d_ws;
  size_t off = 0;
  auto take = [&](size_t bytes) -> void* {
    off = (off + 255) & ~(size_t)255;
    void* p = ws + off; off += bytes; return p;
  };
  float* X    = (float*)take((size_t)MM * DD  * 4);
  float* buf1 = (float*)take((size_t)MM * HID * 4);
  float* buf2 = (float*)take((size_t)MM * HID * 4);
  float* mo   = (float*)take((size_t)MM * 4);
  float* zo   = (float*)take((size_t)MM * 4);
  float* lm_s = (float*)take((size_t)NN * LL * 4);
  float* z1_s = (float*)take((size_t)NN * LL * 4);
  int*   intv  = (int*)take((size_t)NN * 4);
  int*   order = (int*)take((size_t)NN * 4);
  int*   base  = (int*)take((size_t)(NB + 1) * 4);
  int*   alt   = (int*)take((size_t)NN * 4);

  mie_intv<<<NN / 256, 256, 0, stream>>>(target, intv);
  mie_sort<<<1, 32, 0, stream>>>(intv, order, base);
  mie_build<<<NN, 128, 0, stream>>>(z_sample, tparams, gnoise, gprobs, bern,
                                    order, intv, X, lm_s, z1_s);
  mie_alt<<<NN / 256, 256, 0, stream>>>(pnoise, base, order, intv, alt);
  mie_fill_alt<<<NN, 128, 0, stream>>>(alt, lm_s, z1_s, X);

  dim3 gemm_grid(HID / BN, MM / BM);   // (8, 256)
  auto run_mlp = [&](const float* const* w, float* outv) {
    mie_gemm<<<gemm_grid, 256, 0, stream>>>(X,    w[0], buf1, MM, HID, DD);
    mie_ln_silu<<<MM, 256, 0, stream>>>(buf1, w[1], w[2], w[3]);
    mie_gemm<<<gemm_grid, 256, 0, stream>>>(buf1, w[4], buf2, MM, HID, HID);
    mie_ln_silu<<<MM, 256, 0, stream>>>(buf2, w[5], w[6], w[7]);
    mie_rowdot<<<MM / 8, 256, 0, stream>>>(buf2, w[8], outv);
  };
  run_mlp(miw, mo);   // model_out (stop_gradient only affects grads)
  run_mlp(exw, zo);   // z_out

  mie_loss<<<1, 256, 0, stream>>>(mo, zo, (float*)d_out);
}